// EncoderGIN_62414464745851
// MI455X (gfx1250) — compile-verified
//
#include <hip/hip_runtime.h>
#include <hip/hip_bf16.h>

// ---------------------------------------------------------------------------
// GIN encoder forward, MI455X (gfx1250, wave32).
//   per layer: agg = scatter_add(h[src] -> dst); z = h + agg
//              z = relu(z@W1+b1); z = relu(z@W2+b2); batchnorm; relu
// GEMMs: V_WMMA_F32_16X16X4_F32 (exact fp32). A-tile and transposed W staged
// in LDS (conflict-free ds_load_b64 fragments, h+agg fused during staging).
// Scatter: global_atomic_add_f32 (node features L2-resident, 51 MB << 192 MB).
// BN stats fused into GEMM2 epilogue (shfl_xor 16 + per-column L2 atomics).
// ---------------------------------------------------------------------------

typedef __attribute__((ext_vector_type(2))) float v2f;
typedef __attribute__((ext_vector_type(8))) float v8f;

#define GIN_N      100000
#define GIN_HID    128
#define GIN_INDIM  64
#define GIN_EPS    1e-5f

// ---------------------------------------------------------------- zero fill
__global__ void gin_zero_kernel(float* __restrict__ p, int count4) {
    int i = blockIdx.x * blockDim.x + threadIdx.x;
    if (i < count4) {
        float4 z = {0.f, 0.f, 0.f, 0.f};
        *(float4*)(p + (size_t)i * 4) = z;
    }
}

// ------------------------------------------------------------- scatter-add
// One thread per (edge, 4-channel group): float4 gather from h[src], four
// global_atomic_add_f32 into agg[dst]. Both operands resolve in L2.
__global__ void gin_scatter_kernel(const float* __restrict__ h,
                                   const int* __restrict__ edges,
                                   float* __restrict__ agg,
                                   int n_edges, int dim) {
    int groups = dim >> 2;
    long long tid = (long long)blockIdx.x * blockDim.x + threadIdx.x;
    long long total = (long long)n_edges * groups;
    if (tid >= total) return;
    int e = (int)(tid / groups);
    int g = (int)(tid - (long long)e * groups);
    int src = edges[e];
    int dst = edges[n_edges + e];
    const float4 v = *(const float4*)(h + (size_t)src * dim + g * 4);
    float* d = agg + (size_t)dst * dim + g * 4;
    atomicAdd(d + 0, v.x);
    atomicAdd(d + 1, v.y);
    atomicAdd(d + 2, v.z);
    atomicAdd(d + 3, v.w);
}

// ----------------------------------------------------------------- WMMA GEMM
// out[N x 128] = relu( (A [+ Aagg]) [N x K] @ W [K x 128] + bias )
// Block = 256 thr = 8 waves; block owns 80 rows (N = 1250*80 exact, no tail,
// EXEC all-ones for WMMA). Wave w owns cols [16w,16w+16), 5 row-tiles each.
// W staged transposed in LDS (stride K+4): B fragment = one ds_load_b64,
// banks (4c+kb)%64 -> conflict-free. A tile staged likewise with h+agg fused.
template <int K, bool HAS_AGG, bool HAS_STATS>
__global__ __launch_bounds__(256)
void gin_gemm_kernel(const float* __restrict__ A,
                     const float* __restrict__ Aagg,
                     const float* __restrict__ W,      // [K x 128] row-major
                     const float* __restrict__ bias,   // [128]
                     float* __restrict__ out,          // [N x 128]
                     float* __restrict__ stats) {      // [0..127]=sum,[128..255]=sumsq
    constexpr int LDK  = K + 4;   // padded k-stride (floats)
    constexpr int MT   = 5;       // row tiles per wave
    constexpr int ROWS = 16 * MT; // 80 rows per block

    __shared__ float sWt[GIN_HID * LDK]; // W transposed: [col][k]
    __shared__ float sA[ROWS * LDK];     // (h + agg) tile: [row][k]

    const int tid = threadIdx.x;
    const int R0  = blockIdx.x * ROWS;

    // Stage W transposed (coalesced float4 global read, scalar LDS writes).
    for (int i = tid * 4; i < K * GIN_HID; i += 256 * 4) {
        int k = i >> 7;        // / 128
        int c = i & 127;
        float4 w4 = *(const float4*)(W + (size_t)k * GIN_HID + c);
        sWt[(c + 0) * LDK + k] = w4.x;
        sWt[(c + 1) * LDK + k] = w4.y;
        sWt[(c + 2) * LDK + k] = w4.z;
        sWt[(c + 3) * LDK + k] = w4.w;
    }
    // Stage A tile; fuse the aggregation add here (once, not per column tile).
    for (int i = tid * 4; i < ROWS * K; i += 256 * 4) {
        int r = i / K;
        int c = i - r * K;
        float4 a4 = *(const float4*)(A + (size_t)(R0 + r) * K + c);
        if (HAS_AGG) {
            float4 g4 = *(const float4*)(Aagg + (size_t)(R0 + r) * K + c);
            a4.x += g4.x; a4.y += g4.y; a4.z += g4.z; a4.w += g4.w;
        }
        *(float4*)(sA + r * LDK + c) = a4;  // LDK%4==0, c%4==0 -> 16B aligned
    }
    __syncthreads();

    const int lane    = tid & 31;
    const int wave    = tid >> 5;
    const int halfsel = lane >> 4;      // 0: lanes 0-15, 1: lanes 16-31
    const int l15     = lane & 15;
    const int col     = wave * 16 + l15;

    v8f acc[MT] = {};
    for (int k0 = 0; k0 < K; k0 += 4) {
        const int kb = k0 + (halfsel << 1);                 // lane's K pair
        v2f b = *(const v2f*)(sWt + col * LDK + kb);        // ds_load_b64
#pragma unroll
        for (int mt = 0; mt < MT; ++mt) {
            v2f a = *(const v2f*)(sA + (mt * 16 + l15) * LDK + kb);
            acc[mt] = __builtin_amdgcn_wmma_f32_16x16x4_f32(
                /*neg_a=*/false, a, /*neg_b=*/false, b,
                /*c_mod=*/(short)0, acc[mt],
                /*reuse_a=*/false, /*reuse_b=*/false);
        }
    }

    // Epilogue: bias + relu + store; BN statistics if requested.
    const float bv = bias[col];
    float lsum = 0.0f, lsq = 0.0f;
#pragma unroll
    for (int mt = 0; mt < MT; ++mt) {
#pragma unroll
        for (int r = 0; r < 8; ++r) {
            const int row = R0 + mt * 16 + r + (halfsel << 3);
            float v = acc[mt][r] + bv;
            v = v > 0.0f ? v : 0.0f;
            out[(size_t)row * GIN_HID + col] = v;
            lsum += v;
            lsq  += v * v;
        }
    }
    if (HAS_STATS) {
        // Lanes l and l+16 share a column: fold across wave32 halves.
        lsum += __shfl_xor(lsum, 16, 32);
        lsq  += __shfl_xor(lsq, 16, 32);
        if (lane < 16) {
            atomicAdd(&stats[col], lsum);
            atomicAdd(&stats[GIN_HID + col], lsq);
        }
    }
}

// -------------------------------------------------------- BatchNorm + ReLU
__global__ void gin_bn_relu_kernel(const float* __restrict__ z,
                                   const float* __restrict__ stats,
                                   const float* __restrict__ gamma,
                                   const float* __restrict__ beta,
                                   float* __restrict__ out) {
    int i = blockIdx.x * blockDim.x + threadIdx.x;           // float4 index
    if (i >= (GIN_N * GIN_HID) / 4) return;
    int c0 = (i * 4) & (GIN_HID - 1);
    const float inv_n = 1.0f / (float)GIN_N;
    float4 zv = *(const float4*)(z + (size_t)i * 4);
    float4 ov;
    float* zp = &zv.x;
    float* op = &ov.x;
#pragma unroll
    for (int j = 0; j < 4; ++j) {
        int c = c0 + j;
        float mean = stats[c] * inv_n;
        float var  = stats[GIN_HID + c] * inv_n - mean * mean;
        float rstd = rsqrtf(var + GIN_EPS);
        float v = (zp[j] - mean) * rstd * gamma[c] + beta[c];
        op[j] = v > 0.0f ? v : 0.0f;
    }
    *(float4*)(out + (size_t)i * 4) = ov;
}

// ---------------------------------------------------------------------------
extern "C" void kernel_launch(void* const* d_in, const int* in_sizes, int n_in,
                              void* d_out, int out_size, void* d_ws, size_t ws_size,
                              hipStream_t stream) {
    const float* x     = (const float*)d_in[0];
    const int*   edges = (const int*)d_in[1];
    const int n_edges  = in_sizes[1] / 2;

    // Workspace: 3 rotating N x 128 fp32 buffers + 256-float stats.
    float* buf0  = (float*)d_ws;                         // h (layer in/out)
    float* buf1  = buf0 + (size_t)GIN_N * GIN_HID;       // agg, later z2
    float* buf2  = buf1 + (size_t)GIN_N * GIN_HID;       // z1
    float* stats = buf2 + (size_t)GIN_N * GIN_HID;       // [256]

    const dim3 gemm_grid(GIN_N / 80);   // 1250 blocks, exact (no tail)
    const dim3 gemm_block(256);         // 8 waves of 32

    for (int L = 0; L < 3; ++L) {
        const int K = (L == 0) ? GIN_INDIM : GIN_HID;
        const float* h = (L == 0) ? x : buf0;
        float* outL    = (L == 2) ? (float*)d_out : buf0;

        const float* w1    = (const float*)d_in[2 + L * 6 + 0];
        const float* b1    = (const float*)d_in[2 + L * 6 + 1];
        const float* w2    = (const float*)d_in[2 + L * 6 + 2];
        const float* b2    = (const float*)d_in[2 + L * 6 + 3];
        const float* gamma = (const float*)d_in[2 + L * 6 + 4];
        const float* beta  = (const float*)d_in[2 + L * 6 + 5];

        // 1) zero agg buffer and BN stats
        {
            int cnt4 = (GIN_N * K) / 4;
            gin_zero_kernel<<<(cnt4 + 255) / 256, 256, 0, stream>>>(buf1, cnt4);
            gin_zero_kernel<<<1, 64, 0, stream>>>(stats, 64);
        }
        // 2) scatter-add over edges: agg[dst] += h[src]
        {
            long long total = (long long)n_edges * (K >> 2);
            int blocks = (int)((total + 255) / 256);
            gin_scatter_kernel<<<blocks, 256, 0, stream>>>(h, edges, buf1,
                                                           n_edges, K);
        }
        // 3) z1 = relu((h + agg) @ W1 + b1)
        if (L == 0) {
            gin_gemm_kernel<GIN_INDIM, true, false>
                <<<gemm_grid, gemm_block, 0, stream>>>(h, buf1, w1, b1, buf2,
                                                       nullptr);
        } else {
            gin_gemm_kernel<GIN_HID, true, false>
                <<<gemm_grid, gemm_block, 0, stream>>>(h, buf1, w1, b1, buf2,
                                                       nullptr);
        }
        // 4) z2 = relu(z1 @ W2 + b2), fused BN statistics
        gin_gemm_kernel<GIN_HID, false, true>
            <<<gemm_grid, gemm_block, 0, stream>>>(buf2, nullptr, w2, b2, buf1,
                                                   stats);
        // 5) h_next = relu(batchnorm(z2))
        {
            int cnt4 = (GIN_N * GIN_HID) / 4;
            gin_bn_relu_kernel<<<(cnt4 + 255) / 256, 256, 0, stream>>>(
                buf1, stats, gamma, beta, outL);
        }
    }
    (void)n_in; (void)out_size; (void)ws_size;
}